// PanopticSegmentationGenerator_16080357556722
// MI455X (gfx1250) — compile-verified
//
#include <hip/hip_runtime.h>

#define OUT_HH 640
#define OUT_WW 640
#define NDET 100
#define NSEM 54
#define NSEG 144   // STUFF_OFFSET(90) + NSEM(54)
#define MH 28
#define MW 28
#define SEG_IN 160

typedef __attribute__((ext_vector_type(2))) float v2f;
typedef __attribute__((ext_vector_type(8))) float v8f;

static __device__ __forceinline__ int imin(int a, int b) { return a < b ? a : b; }
static __device__ __forceinline__ int imax(int a, int b) { return a > b ? a : b; }

// ---------------------------------------------------------------------------
// K0: rank-sort detections by descending score (stable: ties keep index order)
// ---------------------------------------------------------------------------
__global__ void sort_kernel(const float* __restrict__ scores, int* __restrict__ order) {
    int b = blockIdx.x;
    int i = threadIdx.x;
    if (i >= NDET) return;
    const float* s = scores + b * NDET;
    float si = s[i];
    int rank = 0;
    for (int j = 0; j < NDET; ++j) {
        float sj = s[j];
        rank += (sj > si) || (sj == si && j < i);
    }
    order[b * NDET + rank] = i;
}

// ---------------------------------------------------------------------------
// K_init: cat=0, inst=-1, areas=0
// ---------------------------------------------------------------------------
__global__ void init_kernel(float* __restrict__ catp, float* __restrict__ instp,
                            int* __restrict__ areas, int total, int nareas) {
    int i = blockIdx.x * blockDim.x + threadIdx.x;
    if (i < total) { catp[i] = 0.0f; instp[i] = -1.0f; }
    if (i < nareas) areas[i] = 0;
}

// ---------------------------------------------------------------------------
// K1: bilinear resize (160->640, half-pixel, edge clamp) + argmax + remap
// ---------------------------------------------------------------------------
__global__ void seg_kernel(const float* __restrict__ logits, int* __restrict__ seg, int B) {
    int per = OUT_HH * OUT_WW;
    int idx = blockIdx.x * blockDim.x + threadIdx.x;
    if (idx >= B * per) return;
    int b = idx / per;
    int r = idx - b * per;
    int y = r / OUT_WW;
    int x = r - y * OUT_WW;

    float sy = (y + 0.5f) * 0.25f - 0.5f;
    float sx = (x + 0.5f) * 0.25f - 0.5f;
    int y0 = (int)floorf(sy); float fy = sy - (float)y0;
    int x0 = (int)floorf(sx); float fx = sx - (float)x0;
    int yc0 = imin(imax(y0, 0), SEG_IN - 1), yc1 = imin(imax(y0 + 1, 0), SEG_IN - 1);
    int xc0 = imin(imax(x0, 0), SEG_IN - 1), xc1 = imin(imax(x0 + 1, 0), SEG_IN - 1);

    const float* base = logits + (size_t)b * SEG_IN * SEG_IN * NSEM;
    const float* p00 = base + ((size_t)yc0 * SEG_IN + xc0) * NSEM;
    const float* p01 = base + ((size_t)yc0 * SEG_IN + xc1) * NSEM;
    const float* p10 = base + ((size_t)yc1 * SEG_IN + xc0) * NSEM;
    const float* p11 = base + ((size_t)yc1 * SEG_IN + xc1) * NSEM;
    float w00 = (1.0f - fy) * (1.0f - fx), w01 = (1.0f - fy) * fx;
    float w10 = fy * (1.0f - fx),          w11 = fy * fx;

    float best = -3.402823466e38f;
    int bi = 0;
    for (int c = 0; c < NSEM; ++c) {
        float v = w00 * p00[c] + w01 * p01[c] + w10 * p10[c] + w11 * p11[c];
        if (v > best) { best = v; bi = c; }
    }
    seg[idx] = (bi <= 1) ? bi : (bi + 90);  // VOID=0, THINGS=1, stuff +STUFF_OFFSET
}

// ---------------------------------------------------------------------------
// 16x16 pasted-mask tile via 7x V_WMMA_F32_16X16X4_F32:
//   P = Wy(16x28) * M(28x28) * Wx^T(28x16)
// B operand columns are the 2-tap x-interpolation of the LDS mask (M*Wx^T),
// A operand rows are the bilinear y-weights computed on the fly.
// A layout (16x4 f32): lane&15 = M row; vgpr0/1 = K, K+1; lanes>=16 -> K+2,K+3.
// B layout (4x16 f32): lane&15 = N col; same K split across lane halves.
// ---------------------------------------------------------------------------
static __device__ __forceinline__ v8f paste_tile(const float* __restrict__ Mlds,
                                                 int ybase, int xbase,
                                                 float by0, float bx0,
                                                 float hh, float ww, int lane) {
    int m15 = lane & 15;
    int kh  = (lane >> 4) << 1;  // 0 or 2
    float my = ((float)(ybase + m15) + 0.5f - by0) * (28.0f / hh) - 0.5f;
    float mx = ((float)(xbase + m15) + 0.5f - bx0) * (28.0f / ww) - 0.5f;
    int j0 = (int)floorf(mx);
    float fx = mx - (float)j0;
    int j1 = j0 + 1;
    bool j0ok = (j0 >= 0) && (j0 < MW);
    bool j1ok = (j1 >= 0) && (j1 < MW);
    float w0 = j0ok ? (1.0f - fx) : 0.0f;
    float w1 = j1ok ? fx : 0.0f;
    int js0 = j0ok ? j0 : 0;
    int js1 = j1ok ? j1 : 0;

    v8f acc = {};
#pragma unroll
    for (int k0 = 0; k0 < MH; k0 += 4) {
        int ka = k0 + kh, kb = ka + 1;
        float a0 = fmaxf(0.0f, 1.0f - fabsf(my - (float)ka));
        float a1 = fmaxf(0.0f, 1.0f - fabsf(my - (float)kb));
        float b0 = w0 * Mlds[ka * MW + js0] + w1 * Mlds[ka * MW + js1];
        float b1 = w0 * Mlds[kb * MW + js0] + w1 * Mlds[kb * MW + js1];
        v2f Av = {a0, a1};
        v2f Bv = {b0, b1};
        acc = __builtin_amdgcn_wmma_f32_16x16x4_f32(false, Av, false, Bv,
                                                    (short)0, acc, false, false);
    }
    return acc;
}

// ---------------------------------------------------------------------------
// K2: sequential panoptic scan. One workgroup (32 waves) per batch image.
// Claimed-pixel state lives in a 51.2KB LDS bitmask (CDNA5: 320KB/WGP).
// ---------------------------------------------------------------------------
__global__ __launch_bounds__(1024) void scan_kernel(
    const float* __restrict__ boxes, const float* __restrict__ scores,
    const float* __restrict__ classes, const float* __restrict__ masks,
    const int* __restrict__ order, float* __restrict__ catp, float* __restrict__ instp) {
    __shared__ unsigned claimed[(OUT_HH * OUT_WW) / 32];  // bit per pixel
    __shared__ float Mlds[MH * MW];
    __shared__ int redArea, redOv;

    int b = blockIdx.x;
    int tid = threadIdx.x;
    int lane = tid & 31;
    int wave = tid >> 5;
    int nwaves = (int)(blockDim.x >> 5);

    for (int t = tid; t < (OUT_HH * OUT_WW) / 32; t += blockDim.x) claimed[t] = 0u;
    float* catb  = catp  + (size_t)b * OUT_HH * OUT_WW;
    float* instb = instp + (size_t)b * OUT_HH * OUT_WW;
    __syncthreads();

    for (int k = 0; k < NDET; ++k) {
        int idx   = order[b * NDET + k];
        float s   = scores[b * NDET + idx];
        float cid = classes[b * NDET + idx];
        const float* bx = boxes + ((size_t)b * NDET + idx) * 4;
        float by0 = bx[0], bx0 = bx[1], by1 = bx[2], bx1 = bx[3];
        float hh = fmaxf(by1 - by0, 1e-6f);
        float ww = fmaxf(bx1 - bx0, 1e-6f);

        if (tid == 0) { redArea = 0; redOv = 0; }
        const float* msrc = masks + ((size_t)b * NDET + idx) * MH * MW;
        for (int t = tid; t < MH * MW; t += blockDim.x) Mlds[t] = msrc[t];

        // pixel range where paste weights can be nonzero (with slack)
        int iyl = imax(0, (int)floorf(by0 - hh * (1.0f / 56.0f) - 0.5f));
        int iyh = imin(OUT_HH, (int)ceilf(by0 + hh * (28.5f / 28.0f) - 0.5f) + 1);
        int ixl = imax(0, (int)floorf(bx0 - ww * (1.0f / 56.0f) - 0.5f));
        int ixh = imin(OUT_WW, (int)ceilf(bx0 + ww * (28.5f / 28.0f) - 0.5f) + 1);
        int ty0 = iyl >> 4, ty1 = (iyh + 15) >> 4;
        int tx0 = ixl >> 4, tx1 = (ixh + 15) >> 4;
        int nty = imax(0, ty1 - ty0);
        int ntx = imax(0, tx1 - tx0);
        if (iyh <= iyl || ixh <= ixl) { nty = 0; ntx = 0; }
        int ntiles = nty * ntx;
        __syncthreads();  // Mlds/red ready; prev pass2 claims visible

        // pass 1: area + overlap vs claimed bitmask (tile-per-wave, WMMA paste)
        int myArea = 0, myOv = 0;
        int col = lane & 15;
        int rhi = (lane >> 4) << 3;
        for (int t = wave; t < ntiles; t += nwaves) {
            int ty = ty0 + t / ntx;
            int tx = tx0 + t - (t / ntx) * ntx;
            int ybase = ty << 4, xbase = tx << 4;
            v8f acc = paste_tile(Mlds, ybase, xbase, by0, bx0, hh, ww, lane);
#pragma unroll
            for (int v = 0; v < 8; ++v) {
                float P = acc[v];
                if (P > 0.5f) {
                    int p = (ybase + v + rhi) * OUT_WW + (xbase + col);
                    myArea++;
                    myOv += (int)((claimed[p >> 5] >> (p & 31)) & 1u);
                }
            }
        }
        if (myArea) atomicAdd(&redArea, myArea);
        if (myOv) atomicAdd(&redOv, myOv);
        __syncthreads();

        int area = redArea, ov = redOv;
        bool valid = (s > 0.5f) && (area > 0) && (2 * ov <= area);  // OVERLAP_THR=0.5 exact
        if (valid) {
            float fInst = (float)(idx + 1);
            for (int t = wave; t < ntiles; t += nwaves) {
                int ty = ty0 + t / ntx;
                int tx = tx0 + t - (t / ntx) * ntx;
                int ybase = ty << 4, xbase = tx << 4;
                v8f acc = paste_tile(Mlds, ybase, xbase, by0, bx0, hh, ww, lane);
#pragma unroll
                for (int v = 0; v < 8; ++v) {
                    float P = acc[v];
                    if (P > 0.5f) {
                        int p = (ybase + v + rhi) * OUT_WW + (xbase + col);
                        unsigned bit = 1u << (p & 31);
                        unsigned old = atomicOr(&claimed[p >> 5], bit);
                        if (!(old & bit)) { catb[p] = cid; instb[p] = fInst; }
                    }
                }
            }
        }
        __syncthreads();  // pass2 complete before next detection
    }
}

// ---------------------------------------------------------------------------
// K3: per-class void-pixel areas (LDS histogram -> global)
// ---------------------------------------------------------------------------
__global__ void area_kernel(const float* __restrict__ catp, const int* __restrict__ seg,
                            int* __restrict__ areas) {
    __shared__ int hist[NSEG];
    int b = blockIdx.y;
    for (int t = threadIdx.x; t < NSEG; t += blockDim.x) hist[t] = 0;
    __syncthreads();
    int per = OUT_HH * OUT_WW;
    for (int i = blockIdx.x * blockDim.x + threadIdx.x; i < per; i += gridDim.x * blockDim.x) {
        size_t g = (size_t)b * per + i;
        if (catp[g] == 0.0f) atomicAdd(&hist[seg[g]], 1);
    }
    __syncthreads();
    for (int t = threadIdx.x; t < NSEG; t += blockDim.x)
        if (hist[t]) atomicAdd(&areas[b * NSEG + t], hist[t]);
}

// ---------------------------------------------------------------------------
// K4: fill kept stuff segments into cat
// ---------------------------------------------------------------------------
__global__ void stuff_kernel(float* __restrict__ catp, const int* __restrict__ seg,
                             const int* __restrict__ areas, int B) {
    int per = OUT_HH * OUT_WW;
    int i = blockIdx.x * blockDim.x + threadIdx.x;
    if (i >= B * per) return;
    int b = i / per;
    if (catp[i] == 0.0f) {
        int sv = seg[i];
        if (sv != 1 && areas[b * NSEG + sv] >= 4096) catp[i] = (float)sv;  // STUFF_AREA
    }
}

// ---------------------------------------------------------------------------
extern "C" void kernel_launch(void* const* d_in, const int* in_sizes, int n_in,
                              void* d_out, int out_size, void* d_ws, size_t ws_size,
                              hipStream_t stream) {
    const float* boxes   = (const float*)d_in[0];  // (B,100,4)
    const float* scores  = (const float*)d_in[1];  // (B,100)
    const float* classes = (const float*)d_in[2];  // (B,100)
    const float* masks   = (const float*)d_in[3];  // (B,100,28,28)
    const float* logits  = (const float*)d_in[4];  // (B,160,160,54)
    (void)n_in; (void)ws_size; (void)out_size;

    int B = in_sizes[1] / NDET;
    int per = OUT_HH * OUT_WW;
    int total = B * per;

    float* catp  = (float*)d_out;
    float* instp = catp + total;

    int* seg   = (int*)d_ws;          // B*per
    int* order = seg + total;         // B*NDET
    int* areas = order + B * NDET;    // B*NSEG

    init_kernel<<<(total + 255) / 256, 256, 0, stream>>>(catp, instp, areas, total, B * NSEG);
    sort_kernel<<<B, 128, 0, stream>>>(scores, order);
    seg_kernel<<<(total + 255) / 256, 256, 0, stream>>>(logits, seg, B);
    scan_kernel<<<B, 1024, 0, stream>>>(boxes, scores, classes, masks, order, catp, instp);
    area_kernel<<<dim3(64, B), 256, 0, stream>>>(catp, seg, areas);
    stuff_kernel<<<(total + 255) / 256, 256, 0, stream>>>(catp, seg, areas, B);
}